// _YOLO_22522808500297
// MI455X (gfx1250) — compile-verified
//
#include <hip/hip_runtime.h>
#include <hip/hip_bf16.h>

typedef __attribute__((ext_vector_type(2))) float v2f;
typedef __attribute__((ext_vector_type(8))) float v8f;

// Problem constants (from reference): B,H,W,A,C,G
constexpr int Bn = 32, Hn = 52, Wn = 52, An = 5, Cn = 80, Gn = 50;
constexpr int HWn  = Hn * Wn;    // 2704
constexpr int HWAn = HWn * An;   // 13520

// One block per image, 128 threads (4 wave32).
__global__ __launch_bounds__(128) void _YOLO_loss_kernel(
    const float* __restrict__ pred_boxes,   // [B,H,W,A,4] (cx,cy,w,h)
    const float* __restrict__ pred_o,       // [B,H,W,A]
    const float* __restrict__ pred_scores,  // [B,H,W,A,C]
    const float* __restrict__ gt_boxes,     // [B,G,4] (x0,y0,x1,y1)
    const int*   __restrict__ gt_labels,    // [B,G]
    float* __restrict__ out)                // [4]
{
    const int b   = blockIdx.x;
    const int tid = threadIdx.x;

    __shared__ int   sMatch[64];   // matched flat index m = cell*A + best_a (padded)
    __shared__ int   sLbl[64];     // gt label per row (padded)
    __shared__ float sD[1024];     // 4 waves * 8 VGPRs * 32 lanes of D; reused for reductions

    const float* pb_img = pred_boxes  + (size_t)b * HWAn * 4;
    const float* po_img = pred_o      + (size_t)b * HWAn;
    const float* ps_img = pred_scores + (size_t)b * HWAn * Cn;

    float obj_acc = 0.f, noobj_acc = 0.f, bbox_acc = 0.f, clf_acc = 0.f;

    if (tid < 64) { sMatch[tid] = 0; sLbl[tid] = 0; }

    // ---------------- Phase 1: per-GT matching (threads 0..49) ----------------
    if (tid < Gn) {
        const int g = tid;
        const float4 gb = *(const float4*)(gt_boxes + ((size_t)b * Gn + g) * 4);
        // gb = (x0, y0, x1, y1)
        const int cx   = (int)((gb.x + gb.z) * (float)Wn * 0.5f);
        const int cy   = (int)((gb.y + gb.w) * (float)Hn * 0.5f);
        const int cell = cy * Wn + cx;
        const float garea = (gb.z - gb.x) * (gb.w - gb.y);

        float  best_iou = -1.0f;
        int    best_a   = 0;
        float4 best_box = make_float4(0.f, 0.f, 0.f, 0.f);
        for (int a = 0; a < An; ++a) {
            const float4 pbx = *(const float4*)(pb_img + (size_t)(cell * An + a) * 4);
            const float px0 = pbx.x - pbx.z * 0.5f;
            const float py0 = pbx.y - pbx.w * 0.5f;
            const float px1 = pbx.x + pbx.z * 0.5f;
            const float py1 = pbx.y + pbx.w * 0.5f;
            const float parea = (px1 - px0) * (py1 - py0);
            const float ltx = fmaxf(px0, gb.x), lty = fmaxf(py0, gb.y);
            const float rbx = fminf(px1, gb.z), rby = fminf(py1, gb.w);
            const float iw = fmaxf(rbx - ltx, 0.f), ih = fmaxf(rby - lty, 0.f);
            const float inter = iw * ih;
            const float iou = inter / (parea + garea - inter);
            if (iou > best_iou) { best_iou = iou; best_a = a; best_box = pbx; } // first-max tiebreak = jnp.argmax
        }
        const int m = cell * An + best_a;
        sMatch[g] = m;
        sLbl[g]   = gt_labels[(size_t)b * Gn + g];

        const float so = po_img[m];
        obj_acc = (so - best_iou) * (so - best_iou);

        float gcx = (gb.x + gb.z) * 0.5f * (float)Wn; gcx -= floorf(gcx);
        float gcy = (gb.y + gb.w) * 0.5f * (float)Hn; gcy -= floorf(gcy);
        float sbx = best_box.x * (float)Wn;           sbx -= floorf(sbx);
        float sby = best_box.y * (float)Hn;           sby -= floorf(sby);
        const float dw = sqrtf(best_box.z) - sqrtf(gb.z - gb.x);
        const float dh = sqrtf(best_box.w) - sqrtf(gb.w - gb.y);
        bbox_acc = (sbx - gcx) * (sbx - gcx) + (sby - gcy) * (sby - gcy) + dw * dw + dh * dh;
    }
    __syncthreads();

    // ---------------- Phase 2: noobj over UNIQUE matched positions ----------------
    if (tid < Gn) {
        const int m = sMatch[tid];
        bool first = true;
        for (int g2 = 0; g2 < tid; ++g2)
            if (sMatch[g2] == m) { first = false; break; }
        if (first) { const float o = po_img[m]; noobj_acc = o * o; }
    }
    __syncthreads();

    // ---------------- Phase 3: clf = trace(E * E^T) via V_WMMA_F32_16X16X4_F32 ----------------
    // E[row,c] = (pred_scores[m[row], c] - onehot(label[row])[c]) * valid(row), rows padded to 64.
    // A-operand (16x4 f32): lane<16 holds (K=0,1) of row=lane, lane>=16 holds (K=2,3) of row=lane-16.
    // Passing the same regs as A and B yields D = E_tile * E_tile^T (Gram); accumulate over K chunks.
    {
        const int wave = tid >> 5;
        const int lane = tid & 31;
        const int row  = wave * 16 + (lane & 15);
        const int koff = (lane & 16) ? 2 : 0;
        const float validf = (row < Gn) ? 1.0f : 0.0f;
        const int mrow = sMatch[row];   // 0 for padded rows -> in-bounds dummy loads
        const int lbl  = sLbl[row];
        const float* rowp = ps_img + (size_t)mrow * Cn + koff;

        v8f acc = {};
        #pragma unroll 4
        for (int k0 = 0; k0 < Cn; k0 += 4) {
            const v2f v = *(const v2f*)(rowp + k0);   // 8B-aligned: (m*80 + even) floats
            const int kk = k0 + koff;
            v2f ab;
            ab.x = (v.x - ((kk     == lbl) ? 1.0f : 0.0f)) * validf;
            ab.y = (v.y - ((kk + 1 == lbl) ? 1.0f : 0.0f)) * validf;
            acc = __builtin_amdgcn_wmma_f32_16x16x4_f32(
                /*neg_a=*/false, ab, /*neg_b=*/false, ab,
                /*c_mod=*/(short)0, acc, /*reuse_a=*/false, /*reuse_b=*/false);
        }
        #pragma unroll
        for (int r = 0; r < 8; ++r)
            sD[wave * 256 + r * 32 + lane] = acc[r];
    }
    __syncthreads();

    // Extract the 16 diagonal elements per wave tile:
    // diag(i): i<8 -> VGPR i, lane i ; i>=8 -> VGPR i-8, lane i+16.
    if (tid < 64) {
        const int w  = tid >> 4;
        const int i  = tid & 15;
        const int r  = i & 7;
        const int ln = (i < 8) ? i : (i + 16);
        clf_acc = sD[w * 256 + r * 32 + ln];
    }

    // ---------------- Phase 4: block reduction + global float atomics ----------------
    const float vals[4]   = { obj_acc, noobj_acc, bbox_acc, clf_acc };
    const float scales[4] = { 1.0f / Bn, 0.5f / Bn, 5.0f / Bn, 1.0f / Bn }; // /B, LAMBDA_NOOBJ/B, LAMBDA_COORDS/B, /B
    for (int q = 0; q < 4; ++q) {
        __syncthreads();
        sD[tid] = vals[q];
        __syncthreads();
        for (int off = 64; off > 0; off >>= 1) {
            if (tid < off) sD[tid] += sD[tid + off];
            __syncthreads();
        }
        if (tid == 0) atomicAdd(out + q, sD[0] * scales[q]);
    }
}

extern "C" void kernel_launch(void* const* d_in, const int* in_sizes, int n_in,
                              void* d_out, int out_size, void* d_ws, size_t ws_size,
                              hipStream_t stream) {
    const float* pred_boxes  = (const float*)d_in[0];
    const float* pred_o      = (const float*)d_in[1];
    const float* pred_scores = (const float*)d_in[2];
    const float* gt_boxes    = (const float*)d_in[3];
    const int*   gt_labels   = (const int*)d_in[4];
    float* out = (float*)d_out;

    hipMemsetAsync(out, 0, 4 * sizeof(float), stream);   // graph-capture safe
    _YOLO_loss_kernel<<<dim3(Bn), dim3(128), 0, stream>>>(
        pred_boxes, pred_o, pred_scores, gt_boxes, gt_labels, out);

    (void)in_sizes; (void)n_in; (void)out_size; (void)d_ws; (void)ws_size;
}